// GCN_57604101373966
// MI455X (gfx1250) — compile-verified
//
#include <hip/hip_runtime.h>
#include <hip/hip_bf16.h>
#include <math.h>

typedef __attribute__((ext_vector_type(2))) float v2f;
typedef __attribute__((ext_vector_type(8))) float v8f;
typedef __attribute__((ext_vector_type(4))) float f4v;

#define TPB 256

// ---------------- degree / normalization ----------------

__global__ __launch_bounds__(TPB) void k_zero_u32(unsigned* __restrict__ p, int n) {
  int i = blockIdx.x * TPB + threadIdx.x;
  if (i < n) p[i] = 0u;
}

__global__ __launch_bounds__(TPB) void k_count_deg(const long long* __restrict__ col,
                                                   unsigned* __restrict__ deg, int E) {
  int e = blockIdx.x * TPB + threadIdx.x;
  if (e < E) atomicAdd(&deg[(int)col[e]], 1u);
}

__global__ __launch_bounds__(TPB) void k_dinv(const unsigned* __restrict__ deg,
                                              float* __restrict__ dinv, int n) {
  int i = blockIdx.x * TPB + threadIdx.x;
  if (i < n) dinv[i] = rsqrtf((float)(deg[i] + 1u));  // +1 = self loop
}

// ---------------- two-level exclusive scan (builds CSR rowptr) ----------------

__global__ __launch_bounds__(TPB) void k_scan_block(const unsigned* __restrict__ in,
                                                    unsigned* __restrict__ outExc,
                                                    unsigned* __restrict__ blockSums, int n) {
  __shared__ unsigned s[TPB];
  int i = blockIdx.x * TPB + threadIdx.x;
  unsigned v = (i < n) ? in[i] : 0u;
  s[threadIdx.x] = v;
  __syncthreads();
  for (int d = 1; d < TPB; d <<= 1) {
    unsigned t = (threadIdx.x >= d) ? s[threadIdx.x - d] : 0u;
    __syncthreads();
    s[threadIdx.x] += t;
    __syncthreads();
  }
  if (i < n) outExc[i] = s[threadIdx.x] - v;  // exclusive within block
  if (threadIdx.x == TPB - 1) blockSums[blockIdx.x] = s[TPB - 1];
}

__global__ __launch_bounds__(512) void k_scan_sums(unsigned* __restrict__ sums, int nb) {
  __shared__ unsigned s[512];
  int t = threadIdx.x;
  unsigned v = (t < nb) ? sums[t] : 0u;
  s[t] = v;
  __syncthreads();
  for (int d = 1; d < 512; d <<= 1) {
    unsigned x = (t >= d) ? s[t - d] : 0u;
    __syncthreads();
    s[t] += x;
    __syncthreads();
  }
  if (t < nb) sums[t] = s[t] - v;  // exclusive block offsets
}

__global__ __launch_bounds__(TPB) void k_scan_add(unsigned* __restrict__ rowptr,
                                                  unsigned* __restrict__ cursor,
                                                  const unsigned* __restrict__ sums, int n) {
  int i = blockIdx.x * TPB + threadIdx.x;
  if (i < n) {
    unsigned v = rowptr[i] + sums[blockIdx.x];
    rowptr[i] = v;
    cursor[i] = v;
  }
}

__global__ void k_set_last(unsigned* __restrict__ rowptr, int NN, unsigned E) {
  rowptr[NN] = E;
}

__global__ __launch_bounds__(TPB) void k_fill(const long long* __restrict__ row,
                                              const long long* __restrict__ col,
                                              unsigned* __restrict__ cursor,
                                              unsigned* __restrict__ srcidx, int E) {
  int e = blockIdx.x * TPB + threadIdx.x;
  if (e < E) {
    unsigned p = atomicAdd(&cursor[(int)col[e]], 1u);
    srcidx[p] = (unsigned)row[e];
  }
}

// ---------------- fp32 WMMA GEMM: out[M,NOUT] = A[M,K] @ W[K,NOUT] ----------------
// Block = 8 waves, 64 rows (4 row-subtiles) per block; wave w handles columns
// [16w,16w+16) for all 4 subtiles -> each B fragment reused by 4 WMMAs, W L2
// traffic cut 4x vs one-tile blocks.

template <int K, int NOUT, int ROWT>
__global__ __launch_bounds__(TPB) void k_gemm_wmma(const float* __restrict__ A,
                                                   const float* __restrict__ W,
                                                   float* __restrict__ out, int M) {
  constexpr int PA = K + 4;  // conflict-free pitch (stride 4 banks/row)
  __shared__ float ldsA[16 * ROWT * PA];
  const int r0 = blockIdx.x * (16 * ROWT);
  constexpr int KF4 = K / 4;
  for (int i = threadIdx.x; i < 16 * ROWT * KF4; i += TPB) {
    int r = i / KF4, c4 = (i - r * KF4) * 4;
    f4v v = {};
    if (r0 + r < M) v = *(const f4v*)(A + (size_t)(r0 + r) * K + c4);
    float* dst = &ldsA[r * PA + c4];
    dst[0] = v.x; dst[1] = v.y; dst[2] = v.z; dst[3] = v.w;
  }
  __syncthreads();
  const int wave = threadIdx.x >> 5, lane = threadIdx.x & 31;
  const int n0 = wave * 16;
  if (n0 >= NOUT) return;
  const int half = lane >> 4, li = lane & 15;
  v8f c[ROWT];
#pragma unroll
  for (int t = 0; t < ROWT; ++t) c[t] = (v8f){};
#pragma unroll
  for (int kk = 0; kk < K; kk += 4) {
    const int ka = kk + 2 * half;
    v2f b;
    b.x = W[(size_t)ka * NOUT + n0 + li];
    b.y = W[(size_t)(ka + 1) * NOUT + n0 + li];
#pragma unroll
    for (int t = 0; t < ROWT; ++t) {
      const float* ap = &ldsA[(t * 16 + li) * PA + ka];
      v2f a;
      a.x = ap[0]; a.y = ap[1];
      c[t] = __builtin_amdgcn_wmma_f32_16x16x4_f32(false, a, false, b, (short)0, c[t], false, false);
    }
  }
#pragma unroll
  for (int t = 0; t < ROWT; ++t)
#pragma unroll
    for (int v = 0; v < 8; ++v) {
      int m = r0 + t * 16 + v + 8 * half;
      if (m < M) out[(size_t)m * NOUT + n0 + li] = c[v > 7 ? 0 : v, t][v];
    }
}

// ---------------- classifier GEMM: K=64, NOUT=40 via LDS-padded W (no branches) ----

__global__ __launch_bounds__(TPB) void k_gemm_cls(const float* __restrict__ A,
                                                  const float* __restrict__ W,
                                                  const float* __restrict__ bias,
                                                  float* __restrict__ out, int M) {
  constexpr int K = 64, NOUT = 40, NP = 48, PA = K + 4;
  __shared__ float ldsA[16 * PA];
  __shared__ float ldsW[K * NP];  // zero-padded 64x48
  const int r0 = blockIdx.x * 16;
  const int tid = threadIdx.x;
  constexpr int KF4 = K / 4;
  for (int i = tid; i < 16 * KF4; i += TPB) {
    int r = i / KF4, c4 = (i - r * KF4) * 4;
    f4v v = {};
    if (r0 + r < M) v = *(const f4v*)(A + (size_t)(r0 + r) * K + c4);
    float* dst = &ldsA[r * PA + c4];
    dst[0] = v.x; dst[1] = v.y; dst[2] = v.z; dst[3] = v.w;
  }
  for (int i = tid; i < K * NP; i += TPB) {
    int k = i / NP, n = i - k * NP;
    ldsW[i] = (n < NOUT) ? W[k * NOUT + n] : 0.0f;
  }
  __syncthreads();
  const int wave = tid >> 5, lane = tid & 31;
  const int n0 = wave * 16;
  if (n0 >= NP) return;  // waves 0..2 cover the 40 cols
  const int half = lane >> 4, li = lane & 15;
  const int n = n0 + li;
  v8f c = {};
#pragma unroll
  for (int kk = 0; kk < K; kk += 4) {
    const int ka = kk + 2 * half;
    v2f a, b;
    const float* ap = &ldsA[li * PA + ka];
    a.x = ap[0]; a.y = ap[1];
    b.x = ldsW[ka * NP + n];
    b.y = ldsW[(ka + 1) * NP + n];
    c = __builtin_amdgcn_wmma_f32_16x16x4_f32(false, a, false, b, (short)0, c, false, false);
  }
  if (n >= NOUT) return;
  const float bb = bias[n];
#pragma unroll
  for (int v = 0; v < 8; ++v) {
    int m = r0 + v + 8 * half;
    if (m < M) out[(size_t)m * NOUT + n] = c[v] + bb;
  }
}

// ---------------- CSR gather aggregation (pull; zero feature atomics) ----------------
// One wave per node: acc = dinv[n]^2*h[n] + sum_edges dinv[n]*dinv[r]*h[r];
// fused out = tanh(acc + bias). h buffers are L2-resident (2x51MB < 192MB).

template <int F>
__global__ __launch_bounds__(TPB) void k_gather(const float* __restrict__ h,
                                                const unsigned* __restrict__ rowptr,
                                                const unsigned* __restrict__ srcidx,
                                                const float* __restrict__ dinv,
                                                const float* __restrict__ bias,
                                                float* __restrict__ out, int NN) {
  constexpr int FV = F / 32;  // floats per lane (4 or 2)
  int node = blockIdx.x * (TPB / 32) + (threadIdx.x >> 5);
  if (node >= NN) return;
  const int lane = threadIdx.x & 31;
  const float dn = dinv[node];
  float acc[FV];
  {
    const float* hp = h + (size_t)node * F + lane * FV;
    if constexpr (FV == 4) {
      f4v v = *(const f4v*)hp;
      acc[0] = dn * dn * v.x; acc[1] = dn * dn * v.y;
      acc[2] = dn * dn * v.z; acc[3] = dn * dn * v.w;
    } else {
      v2f v = *(const v2f*)hp;
      acc[0] = dn * dn * v.x; acc[1] = dn * dn * v.y;
    }
  }
  const unsigned beg = rowptr[node], end = rowptr[node + 1];
  for (unsigned p = beg; p < end; ++p) {
    int r = (int)srcidx[p];
    r = __builtin_amdgcn_readfirstlane(r);  // uniform -> scalar addressing
    const float w = dn * dinv[r];
    const float* hr = h + (size_t)r * F + lane * FV;
    if constexpr (FV == 4) {
      f4v v = *(const f4v*)hr;
      acc[0] += w * v.x; acc[1] += w * v.y; acc[2] += w * v.z; acc[3] += w * v.w;
    } else {
      v2f v = *(const v2f*)hr;
      acc[0] += w * v.x; acc[1] += w * v.y;
    }
  }
  float* op = out + (size_t)node * F + lane * FV;
  const float* bp = bias + lane * FV;
#pragma unroll
  for (int j = 0; j < FV; ++j) op[j] = tanhf(acc[j] + bp[j]);
}

// ---------------- driver ----------------

extern "C" void kernel_launch(void* const* d_in, const int* in_sizes, int n_in,
                              void* d_out, int out_size, void* d_ws, size_t ws_size,
                              hipStream_t stream) {
  const float*     x  = (const float*)d_in[0];
  const long long* ei = (const long long*)d_in[1];  // int64 per reference
  const float* W1 = (const float*)d_in[2];
  const float* b1 = (const float*)d_in[3];
  const float* W2 = (const float*)d_in[4];
  const float* b2 = (const float*)d_in[5];
  const float* W3 = (const float*)d_in[6];
  const float* b3 = (const float*)d_in[7];
  const float* Wc = (const float*)d_in[8];
  const float* bc = (const float*)d_in[9];

  const int NN = in_sizes[0] / 128;  // nodes
  const int E  = in_sizes[1] / 2;    // edges

  char* ws = (char*)d_ws;
  size_t off = 0;
  auto take = [&](size_t bytes) {
    char* p = ws + off;
    off = (off + bytes + 255) & ~(size_t)255;
    return p;
  };
  float*    dinv   = (float*)take((size_t)NN * 4);
  unsigned* deg    = (unsigned*)take((size_t)NN * 4);
  unsigned* rowptr = (unsigned*)take((size_t)(NN + 1) * 4);
  unsigned* cursor = (unsigned*)take((size_t)NN * 4);
  unsigned* bsums  = (unsigned*)take(512 * 4);
  unsigned* srcidx = (unsigned*)take((size_t)E * 4);
  float*    bufA   = (float*)take((size_t)NN * 128 * 4);
  float*    bufB   = (float*)take((size_t)NN * 128 * 4);

  const int gN = (NN + TPB - 1) / TPB;  // also = number of scan blocks (<=512)
  const int gE = (E + TPB - 1) / TPB;
  const int gT = (NN + 63) / 64;        // 64-row GEMM blocks
  const int gC = (NN + 15) / 16;        // classifier blocks
  const int gG = (NN + (TPB / 32) - 1) / (TPB / 32);  // gather: wave per node

  // normalization + CSR build (once, reused by all 3 layers)
  k_zero_u32<<<gN, TPB, 0, stream>>>(deg, NN);
  k_count_deg<<<gE, TPB, 0, stream>>>(ei + E, deg, E);
  k_dinv<<<gN, TPB, 0, stream>>>(deg, dinv, NN);
  k_scan_block<<<gN, TPB, 0, stream>>>(deg, rowptr, bsums, NN);
  k_scan_sums<<<1, 512, 0, stream>>>(bsums, gN);
  k_scan_add<<<gN, TPB, 0, stream>>>(rowptr, cursor, bsums, NN);
  k_set_last<<<1, 1, 0, stream>>>(rowptr, NN, (unsigned)E);
  k_fill<<<gE, TPB, 0, stream>>>(ei, ei + E, cursor, srcidx, E);

  // layer 1: 128 -> 128
  k_gemm_wmma<128, 128, 4><<<gT, TPB, 0, stream>>>(x, W1, bufA, NN);
  k_gather<128><<<gG, TPB, 0, stream>>>(bufA, rowptr, srcidx, dinv, b1, bufB, NN);
  // layer 2: 128 -> 128
  k_gemm_wmma<128, 128, 4><<<gT, TPB, 0, stream>>>(bufB, W2, bufA, NN);
  k_gather<128><<<gG, TPB, 0, stream>>>(bufA, rowptr, srcidx, dinv, b2, bufB, NN);
  // layer 3: 128 -> 64
  k_gemm_wmma<128, 64, 4><<<gT, TPB, 0, stream>>>(bufB, W3, bufA, NN);
  k_gather<64><<<gG, TPB, 0, stream>>>(bufA, rowptr, srcidx, dinv, b3, bufB, NN);
  // classifier: 64 -> 40 (+bias) straight to d_out
  k_gemm_cls<<<gC, TPB, 0, stream>>>(bufB, Wc, bc, (float*)d_out, NN);
}